// Model_35931696398571
// MI455X (gfx1250) — compile-verified
//
#include <hip/hip_runtime.h>
#include <hip/hip_bf16.h>
#include <math.h>

typedef __attribute__((ext_vector_type(2))) float v2f;
typedef __attribute__((ext_vector_type(8))) float v8f;

#define NPTS 2048
#define BATCH 8
#define KNN 10

// ---------------------------------------------------------------------------
// gfx1250 async global->LDS copy (ASYNCcnt-tracked), GV addressing mode.
// vdst = per-lane LDS byte offset, vaddr = per-lane 64-bit global address.
// ---------------------------------------------------------------------------
__device__ __forceinline__ void async_copy_f32_to_lds(const float* gptr,
                                                      float* lptr) {
  const unsigned lds = (unsigned)(uintptr_t)lptr;  // low 32b of generic = LDS off
  const unsigned long long ga = (unsigned long long)(uintptr_t)gptr;
  asm volatile("global_load_async_to_lds_b32 %0, %1, off" ::"v"(lds), "v"(ga)
               : "memory");
}
__device__ __forceinline__ void async_fence() {
  asm volatile("s_wait_asynccnt 0x0" ::: "memory");
}

// ---------------------------------------------------------------------------
// kNN: one block per (b, i). Distances in LDS, 10 rounds of block argmax with
// lower-index tie-break (matches jax.lax.top_k stable ordering).
// ---------------------------------------------------------------------------
template <int C>
__global__ __launch_bounds__(256) void knn_kernel(const float* __restrict__ X,
                                                  int cStride, int c0,
                                                  int* __restrict__ idxOut) {
  const int b = blockIdx.y;
  const int i = blockIdx.x;
  const int tid = threadIdx.x;
  __shared__ float xi[C];
  __shared__ float dist[NPTS];
  __shared__ float rv[256];
  __shared__ int ri[256];

  const float* Xb = X + ((size_t)b * cStride + c0) * NPTS;
  if (tid < C) xi[tid] = Xb[(size_t)tid * NPTS + i];
  __syncthreads();

  for (int j = tid; j < NPTS; j += 256) {
    float acc = 0.f;
#pragma unroll 8
    for (int c = 0; c < C; ++c) {
      float d = Xb[(size_t)c * NPTS + j] - xi[c];
      acc = fmaf(d, d, acc);
    }
    dist[j] = 0.0f - acc;  // negative squared distance (self == 0, max)
  }
  __syncthreads();

  for (int s = 0; s < KNN; ++s) {
    float bv = -INFINITY;
    int bi = NPTS;
    for (int j = tid; j < NPTS; j += 256) {
      float v = dist[j];
      if (v > bv || (v == bv && j < bi)) { bv = v; bi = j; }
    }
    rv[tid] = bv; ri[tid] = bi;
    __syncthreads();
    for (int off = 128; off > 0; off >>= 1) {
      if (tid < off) {
        float v2 = rv[tid + off]; int i2 = ri[tid + off];
        if (v2 > rv[tid] || (v2 == rv[tid] && i2 < ri[tid])) {
          rv[tid] = v2; ri[tid] = i2;
        }
      }
      __syncthreads();
    }
    if (tid == 0) {
      int sel = ri[0];
      idxOut[((size_t)b * NPTS + i) * KNN + s] = sel;
      dist[sel] = -INFINITY;
    }
    __syncthreads();
  }
}

// ---------------------------------------------------------------------------
// Edge-feature build: F[b, c', n*K+k] ; c'<C -> x[j]-x[n] ; else x[n]
// grid: (M/256, 2C, B)
// ---------------------------------------------------------------------------
template <int C>
__global__ __launch_bounds__(256) void build_feat(const float* __restrict__ X,
                                                  int cStride, int c0,
                                                  const int* __restrict__ idx,
                                                  float* __restrict__ F) {
  const int b = blockIdx.z;
  const int cp = blockIdx.y;
  const int m = blockIdx.x * 256 + threadIdx.x;  // m in [0, N*K)
  const size_t M = (size_t)NPTS * KNN;
  const int n = m / KNN;
  const int k = m - n * KNN;
  const int j = idx[((size_t)b * NPTS + n) * KNN + k];
  const float* Xb = X + ((size_t)b * cStride + c0) * NPTS;
  float v;
  if (cp < C)
    v = Xb[(size_t)cp * NPTS + j] - Xb[(size_t)cp * NPTS + n];
  else
    v = Xb[(size_t)(cp - C) * NPTS + n];
  F[((size_t)b * (2 * C) + cp) * M + m] = v;
}

// ---------------------------------------------------------------------------
// WMMA f32 GEMM + fused BN + mish:  Y[b,o,m] = epi( sum_c W[o,c] * X[b,c,m] )
//   INMODE 0: dense X [B][Cin][M]  (async global->LDS staging)
//   INMODE 1: concat (c<1024 -> glob[b][c] broadcast, else cat[b][c-1024][m])
//   EPIMODE 0: BN + mish ; 1: none
// Block = 256 thr (8 waves). Each wave owns TWO 16x16 D tiles (32 M-cols,
// shared A fragment). 128 O-rows per block. K chunked by 32 channels, staged
// into double-buffered LDS stored column-major ([col][chan], padded row of 34
// floats => conflict-free stores and single aligned ds_load_b64 B-fragments).
// One barrier + one s_wait_asynccnt per chunk; WMMAs overlap next-chunk DMA.
// ---------------------------------------------------------------------------
__device__ __forceinline__ float mish_f(float v) {
  float sp = fmaxf(v, 0.f) + log1pf(expf(-fabsf(v)));  // logaddexp(v, 0)
  return v * tanhf(sp);
}

template <int INMODE, int EPIMODE>
__global__ __launch_bounds__(256) void gemm_wmma(
    const float* __restrict__ W, const float* __restrict__ X,
    const float* __restrict__ G, float* __restrict__ Y, int O, int Cin, int M,
    const float* __restrict__ bng, const float* __restrict__ bnb,
    const float* __restrict__ bnm, const float* __restrict__ bnv) {
  constexpr int KC = 32;
  constexpr int KP = KC + 2;  // pad: stores & b64 reads bank-conflict-free
  const int b = blockIdx.z;
  const int mBase = blockIdx.x * 32;
  const int wave = __builtin_amdgcn_readfirstlane((int)(threadIdx.x >> 5));
  const int lane = threadIdx.x & 31;
  const int half = lane >> 4;
  const int l16 = lane & 15;
  const int oTile = (blockIdx.y * 8 + wave) * 16;
  const bool waveActive = (oTile < O);  // scalar -> uniform branch, EXEC full
  // Clamp A row so main-path W loads need no per-lane guard (clamped rows
  // compute garbage that is never stored).
  const int arow = (oTile + l16 < O) ? (oTile + l16) : (O - 1);

  __shared__ float Xs[2][32][KP];  // [buf][m-col][channel]

  v8f acc0 = {};
  v8f acc1 = {};

  const int nFull = Cin / KC;
  const int tailC = Cin - nFull * KC;
  const int nChunks = nFull + (tailC ? 1 : 0);

  // ---- staging helpers ----------------------------------------------------
  // e -> cc = e>>5 (wave-uniform), col = e&31 (per-lane)
  auto stageFull = [&](int ci, int buf) {
#pragma unroll
    for (int e = threadIdx.x; e < KC * 32; e += 256) {
      const int cc = e >> 5;
      const int col = e & 31;
      const int c = ci * KC + cc;
      if (INMODE == 0) {
        const float* p = &X[((size_t)b * Cin + c) * M + (mBase + col)];
        async_copy_f32_to_lds(p, &Xs[buf][col][cc]);  // DMA, no VGPR round-trip
        if (c + 2 * KC < Cin)  // prefetch chunk ci+2 -> global_prefetch_b8
          __builtin_prefetch(p + (size_t)2 * KC * M, 0, 1);
      } else {
        float val =
            (c < 1024)
                ? G[b * 1024 + c]
                : X[((size_t)b * 192 + (c - 1024)) * (size_t)M + (mBase + col)];
        Xs[buf][col][cc] = val;
      }
    }
  };
  auto stageTail = [&](int ci, int buf) {
#pragma unroll
    for (int e = threadIdx.x; e < KC * 32; e += 256) {
      const int cc = e >> 5;
      const int col = e & 31;
      const int c = ci * KC + cc;
      float val = 0.f;
      if (c < Cin) {
        if (INMODE == 0)
          val = X[((size_t)b * Cin + c) * M + (mBase + col)];
        else
          val = (c < 1024)
                    ? G[b * 1024 + c]
                    : X[((size_t)b * 192 + (c - 1024)) * (size_t)M + (mBase + col)];
      }
      Xs[buf][col][cc] = val;
    }
  };

  // ---- compute helpers ----------------------------------------------------
  auto computeFull = [&](int ci, int buf) {
    const int base = ci * KC;
#pragma unroll
    for (int kk = 0; kk < KC; kk += 4) {
      const int c0 = base + kk + 2 * half;  // even -> 8B-aligned b64 load
      v2f a = *(const v2f*)(W + (size_t)arow * Cin + c0);
      v2f b0 = *(const v2f*)&Xs[buf][l16][kk + 2 * half];
      v2f b1 = *(const v2f*)&Xs[buf][16 + l16][kk + 2 * half];
      acc0 = __builtin_amdgcn_wmma_f32_16x16x4_f32(false, a, false, b0,
                                                   (short)0, acc0, false, false);
      acc1 = __builtin_amdgcn_wmma_f32_16x16x4_f32(false, a, false, b1,
                                                   (short)0, acc1, false, false);
    }
  };
  auto computeTail = [&](int ci, int buf) {
    const int base = ci * KC;
#pragma unroll
    for (int kk = 0; kk < KC; kk += 4) {
      const int c0 = base + kk + 2 * half;
      v2f a;
      a.x = (c0 < Cin) ? W[(size_t)arow * Cin + c0] : 0.f;
      a.y = (c0 + 1 < Cin) ? W[(size_t)arow * Cin + c0 + 1] : 0.f;
      v2f b0 = *(const v2f*)&Xs[buf][l16][kk + 2 * half];
      v2f b1 = *(const v2f*)&Xs[buf][16 + l16][kk + 2 * half];
      acc0 = __builtin_amdgcn_wmma_f32_16x16x4_f32(false, a, false, b0,
                                                   (short)0, acc0, false, false);
      acc1 = __builtin_amdgcn_wmma_f32_16x16x4_f32(false, a, false, b1,
                                                   (short)0, acc1, false, false);
    }
  };

  // ---- K loop, double-buffered, async-overlapped --------------------------
  if (nFull > 0) stageFull(0, 0); else stageTail(0, 0);
  if (INMODE == 0) async_fence();
  __syncthreads();
  for (int ci = 0; ci < nChunks; ++ci) {
    if (ci + 1 < nChunks) {
      if (ci + 1 < nFull) stageFull(ci + 1, (ci + 1) & 1);
      else stageTail(ci + 1, (ci + 1) & 1);
    }
    if (waveActive) {
      if (ci < nFull) computeFull(ci, ci & 1);
      else computeTail(ci, ci & 1);
    }
    if (INMODE == 0) async_fence();  // next-chunk DMA done (overlapped w/ WMMA)
    __syncthreads();
  }

  // ---- epilogue ------------------------------------------------------------
  if (waveActive) {
#pragma unroll
    for (int v = 0; v < 8; ++v) {
      const int row = oTile + v + 8 * half;  // D layout: VGPR v, lane half
      if (row < O) {
        float y0 = acc0[v];
        float y1 = acc1[v];
        if (EPIMODE == 0) {
          const float sc = bng[row] / sqrtf(bnv[row] + 1e-5f);
          const float mu = bnm[row];
          const float bt = bnb[row];
          y0 = mish_f((y0 - mu) * sc + bt);
          y1 = mish_f((y1 - mu) * sc + bt);
        }
        float* yp = Y + ((size_t)b * O + row) * M + mBase + l16;
        yp[0] = y0;
        yp[16] = y1;
      }
    }
  }
}

// ---------------------------------------------------------------------------
// max over K neighbors: H[B][64][N*K] -> cat[b][cOff+c][n]
// ---------------------------------------------------------------------------
__global__ __launch_bounds__(256) void maxk_kernel(const float* __restrict__ H,
                                                   float* __restrict__ cat,
                                                   int cOff) {
  const int n = blockIdx.x * 256 + threadIdx.x;
  const int c = blockIdx.y;
  const int b = blockIdx.z;
  if (n >= NPTS) return;
  const float* p = H + ((size_t)(b * 64 + c)) * NPTS * KNN + (size_t)n * KNN;
  float m = p[0];
#pragma unroll
  for (int k = 1; k < KNN; ++k) m = fmaxf(m, p[k]);
  cat[((size_t)b * 192 + cOff + c) * NPTS + n] = m;
}

// ---------------------------------------------------------------------------
// global max over points: h6[B][1024][N] -> glob[B][1024]
// ---------------------------------------------------------------------------
__global__ __launch_bounds__(256) void globmax_kernel(
    const float* __restrict__ H6, float* __restrict__ Gl) {
  const int c = blockIdx.x;
  const int b = blockIdx.y;
  __shared__ float red[256];
  const float* p = H6 + ((size_t)b * 1024 + c) * NPTS;
  float m = -INFINITY;
  for (int n = threadIdx.x; n < NPTS; n += 256) m = fmaxf(m, p[n]);
  red[threadIdx.x] = m;
  __syncthreads();
  for (int off = 128; off > 0; off >>= 1) {
    if (threadIdx.x < off)
      red[threadIdx.x] = fmaxf(red[threadIdx.x], red[threadIdx.x + off]);
    __syncthreads();
  }
  if (threadIdx.x == 0) Gl[b * 1024 + c] = red[0];
}

// ---------------------------------------------------------------------------
extern "C" void kernel_launch(void* const* d_in, const int* in_sizes, int n_in,
                              void* d_out, int out_size, void* d_ws,
                              size_t ws_size, hipStream_t stream) {
  (void)in_sizes; (void)n_in; (void)out_size; (void)ws_size;
  const float* x = (const float*)d_in[0];
  const float* w1 = (const float*)d_in[1];
  const float* w2 = (const float*)d_in[2];
  const float* w3 = (const float*)d_in[3];
  const float* w4 = (const float*)d_in[4];
  const float* w5 = (const float*)d_in[5];
  const float* w6 = (const float*)d_in[6];
  const float* w7 = (const float*)d_in[7];
  const float* w8 = (const float*)d_in[8];
  const float* w9 = (const float*)d_in[9];
  const float* g15 = (const float*)d_in[10];
  const float* b15 = (const float*)d_in[11];
  const float* m15 = (const float*)d_in[12];
  const float* v15 = (const float*)d_in[13];
  const float* g6 = (const float*)d_in[14];
  const float* b6 = (const float*)d_in[15];
  const float* m6 = (const float*)d_in[16];
  const float* v6 = (const float*)d_in[17];
  const float* g7 = (const float*)d_in[18];
  const float* b7 = (const float*)d_in[19];
  const float* m7 = (const float*)d_in[20];
  const float* v7 = (const float*)d_in[21];
  const float* g8 = (const float*)d_in[22];
  const float* b8 = (const float*)d_in[23];
  const float* m8 = (const float*)d_in[24];
  const float* v8 = (const float*)d_in[25];
  float* out = (float*)d_out;

  const size_t M1 = (size_t)NPTS * KNN;             // 20480
  const size_t FEAT_E = (size_t)BATCH * 128 * M1;   // 20,971,520
  const size_t HA_E   = (size_t)BATCH * 64 * M1;    // 10,485,760
  const size_t CAT_E  = (size_t)BATCH * 192 * NPTS; //  3,145,728
  const size_t H6_E   = (size_t)BATCH * 1024 * NPTS;
  const size_t GL_E   = (size_t)BATCH * 1024;
  const size_t H7_E   = (size_t)BATCH * 512 * NPTS;
  const size_t H8_E   = (size_t)BATCH * 256 * NPTS;

  float* feat = (float*)d_ws;  // also reused as 2nd-conv output
  float* hA   = feat + FEAT_E;
  float* cat  = hA + HA_E;
  float* h6   = cat + CAT_E;
  float* glob = h6 + H6_E;
  float* h7   = glob + GL_E;
  float* h8   = h7 + H7_E;
  int* idx    = (int*)(h8 + H8_E);

  dim3 blk(256);
  const int mt1 = (int)(M1 / 32);  // 640  (32 M-cols per block)
  const int mtp = NPTS / 32;       // 64

  // ---- stage 1 (C=3) ----
  knn_kernel<3><<<dim3(NPTS, BATCH), blk, 0, stream>>>(x, 3, 0, idx);
  build_feat<3><<<dim3((int)(M1 / 256), 6, BATCH), blk, 0, stream>>>(x, 3, 0, idx, feat);
  gemm_wmma<0, 0><<<dim3(mt1, 1, BATCH), blk, 0, stream>>>(
      w1, feat, nullptr, hA, 64, 6, (int)M1, g15 + 0, b15 + 0, m15 + 0, v15 + 0);
  gemm_wmma<0, 0><<<dim3(mt1, 1, BATCH), blk, 0, stream>>>(
      w2, hA, nullptr, feat, 64, 64, (int)M1, g15 + 64, b15 + 64, m15 + 64, v15 + 64);
  maxk_kernel<<<dim3(NPTS / 256, 64, BATCH), blk, 0, stream>>>(feat, cat, 0);

  // ---- stage 2 (C=64, x1 = cat[:,0:64]) ----
  knn_kernel<64><<<dim3(NPTS, BATCH), blk, 0, stream>>>(cat, 192, 0, idx);
  build_feat<64><<<dim3((int)(M1 / 256), 128, BATCH), blk, 0, stream>>>(cat, 192, 0, idx, feat);
  gemm_wmma<0, 0><<<dim3(mt1, 1, BATCH), blk, 0, stream>>>(
      w3, feat, nullptr, hA, 64, 128, (int)M1, g15 + 128, b15 + 128, m15 + 128, v15 + 128);
  gemm_wmma<0, 0><<<dim3(mt1, 1, BATCH), blk, 0, stream>>>(
      w4, hA, nullptr, feat, 64, 64, (int)M1, g15 + 192, b15 + 192, m15 + 192, v15 + 192);
  maxk_kernel<<<dim3(NPTS / 256, 64, BATCH), blk, 0, stream>>>(feat, cat, 64);

  // ---- stage 3 (C=64, x2 = cat[:,64:128]) ----
  knn_kernel<64><<<dim3(NPTS, BATCH), blk, 0, stream>>>(cat, 192, 64, idx);
  build_feat<64><<<dim3((int)(M1 / 256), 128, BATCH), blk, 0, stream>>>(cat, 192, 64, idx, feat);
  gemm_wmma<0, 0><<<dim3(mt1, 1, BATCH), blk, 0, stream>>>(
      w5, feat, nullptr, hA, 64, 128, (int)M1, g15 + 256, b15 + 256, m15 + 256, v15 + 256);
  maxk_kernel<<<dim3(NPTS / 256, 64, BATCH), blk, 0, stream>>>(hA, cat, 128);

  // ---- pointwise head ----
  gemm_wmma<0, 0><<<dim3(mtp, 8, BATCH), blk, 0, stream>>>(
      w6, cat, nullptr, h6, 1024, 192, NPTS, g6, b6, m6, v6);
  globmax_kernel<<<dim3(1024, BATCH), blk, 0, stream>>>(h6, glob);
  gemm_wmma<1, 0><<<dim3(mtp, 4, BATCH), blk, 0, stream>>>(
      w7, cat, glob, h7, 512, 1216, NPTS, g7, b7, m7, v7);
  gemm_wmma<0, 0><<<dim3(mtp, 2, BATCH), blk, 0, stream>>>(
      w8, h7, nullptr, h8, 256, 512, NPTS, g8, b8, m8, v8);
  gemm_wmma<0, 1><<<dim3(mtp, 1, BATCH), blk, 0, stream>>>(
      w9, h8, nullptr, out, 18, 256, NPTS, nullptr, nullptr, nullptr, nullptr);
}